// GINModel_66022237274356
// MI455X (gfx1250) — compile-verified
//
#include <hip/hip_runtime.h>
#include <hip/hip_bf16.h>

typedef float v2f __attribute__((ext_vector_type(2)));
typedef float v8f __attribute__((ext_vector_type(8)));

#define D 64

// ---------------------------------------------------------------------------
// Scatter: z[dst] += h[src] over all edges (z pre-initialized to h, so the
// GIN self-term h + agg comes out of this step directly). 16 threads/edge,
// each handling 4 channels via float4 read + 4 global fp32 atomics.
// ---------------------------------------------------------------------------
__global__ void gin_scatter(const float* __restrict__ h, float* __restrict__ z,
                            const int* __restrict__ src, const int* __restrict__ dst,
                            int E) {
    unsigned t = blockIdx.x * blockDim.x + threadIdx.x;
    int e = (int)(t >> 4);
    if (e >= E) return;
    int c = (int)(t & 15u) * 4;
    int s = src[e];
    int d = dst[e];
    float4 v = *(const float4*)(h + (long long)s * D + c);
    float* zp = z + (long long)d * D + c;
    atomicAdd(zp + 0, v.x);
    atomicAdd(zp + 1, v.y);
    atomicAdd(zp + 2, v.z);
    atomicAdd(zp + 3, v.w);
}

// ---------------------------------------------------------------------------
// Fused GEMM + bias + optional ReLU: out[r,:] = act(in[r,:] @ W + b), W is
// 64x64 row-major, staged in LDS. One wave = 16 output rows x 64 cols using
// v_wmma_f32_16x16x4_f32 (full fp32). 128 threads = 4 waves = 64 rows/block.
// Safe to run in place (out == in): each wave consumes its 16 A-rows across
// the whole K loop before the epilogue stores to those same rows.
// ---------------------------------------------------------------------------
__global__ void gin_mlp_gemm(const float* __restrict__ in, float* __restrict__ out,
                             const float* __restrict__ W, const float* __restrict__ bias,
                             int nrows, int do_relu) {
    __shared__ float lw[D * D];  // 16 KB weight tile
    int tid = threadIdx.x;
    for (int i = tid; i < D * D; i += 128) lw[i] = W[i];
    __syncthreads();

    int wave = tid >> 5;
    int lane = tid & 31;
    int m0 = blockIdx.x * 64 + wave * 16;
    int mrow = lane & 15;       // A: M index / B,C,D: N index
    int khalf = lane >> 4;      // A/B: K pair select; C/D: M high half

    v8f acc0 = {}, acc1 = {}, acc2 = {}, acc3 = {};

    int arow = m0 + mrow;
    int arow_c = arow < nrows ? arow : 0;  // clamp address; zero the value
    const float* arp = in + (long long)arow_c * D;

    for (int k = 0; k < D; k += 4) {
        int kb = k + khalf * 2;
        float a0 = arp[kb];
        float a1 = arp[kb + 1];
        if (arow >= nrows) { a0 = 0.0f; a1 = 0.0f; }
        v2f a = {a0, a1};

        v2f b0 = {lw[kb * D +  0 + mrow], lw[(kb + 1) * D +  0 + mrow]};
        v2f b1 = {lw[kb * D + 16 + mrow], lw[(kb + 1) * D + 16 + mrow]};
        v2f b2 = {lw[kb * D + 32 + mrow], lw[(kb + 1) * D + 32 + mrow]};
        v2f b3 = {lw[kb * D + 48 + mrow], lw[(kb + 1) * D + 48 + mrow]};

        acc0 = __builtin_amdgcn_wmma_f32_16x16x4_f32(false, a, false, b0, (short)0, acc0, false, false);
        acc1 = __builtin_amdgcn_wmma_f32_16x16x4_f32(false, a, false, b1, (short)0, acc1, false, false);
        acc2 = __builtin_amdgcn_wmma_f32_16x16x4_f32(false, a, false, b2, (short)0, acc2, false, false);
        acc3 = __builtin_amdgcn_wmma_f32_16x16x4_f32(false, a, false, b3, (short)0, acc3, false, false);
    }

    // Epilogue: C/D layout -> lane (L&15)=N, VGPR v: M = v + 8*(L>>4)
    v8f accs[4] = {acc0, acc1, acc2, acc3};
#pragma unroll
    for (int nt = 0; nt < 4; ++nt) {
        int col = nt * 16 + mrow;
        float bb = bias[col];
#pragma unroll
        for (int v = 0; v < 8; ++v) {
            int row = m0 + v + khalf * 8;
            if (row < nrows) {
                float val = accs[nt][v] + bb;
                if (do_relu) val = fmaxf(val, 0.0f);
                out[(long long)row * D + col] = val;
            }
        }
    }
}

// ---------------------------------------------------------------------------
// Global add-pool: pooled[batch[n]] += h[n]. 16 threads/node, float4 + atomics.
// ---------------------------------------------------------------------------
__global__ void gin_pool(const float* __restrict__ h, const int* __restrict__ batch,
                         float* __restrict__ pooled, int nrows) {
    unsigned t = blockIdx.x * blockDim.x + threadIdx.x;
    int n = (int)(t >> 4);
    if (n >= nrows) return;
    int c = (int)(t & 15u) * 4;
    int g = batch[n];
    float4 v = *(const float4*)(h + (long long)n * D + c);
    float* p = pooled + (long long)g * D + c;
    atomicAdd(p + 0, v.x);
    atomicAdd(p + 1, v.y);
    atomicAdd(p + 2, v.z);
    atomicAdd(p + 3, v.w);
}

// ---------------------------------------------------------------------------
// Final FC: out[g] = pooled[g,:] . fc_w + fc_b   (G x 64 -> G x 1)
// ---------------------------------------------------------------------------
__global__ void gin_fc(const float* __restrict__ pooled, const float* __restrict__ w,
                       const float* __restrict__ b, float* __restrict__ out, int G) {
    int g = blockIdx.x * blockDim.x + threadIdx.x;
    if (g >= G) return;
    float s = b[0];
#pragma unroll
    for (int d = 0; d < D; ++d) s += pooled[(long long)g * D + d] * w[d];
    out[g] = s;
}

extern "C" void kernel_launch(void* const* d_in, const int* in_sizes, int n_in,
                              void* d_out, int out_size, void* d_ws, size_t ws_size,
                              hipStream_t stream) {
    const float* x   = (const float*)d_in[0];
    const int*   ei  = (const int*)d_in[1];     // [2, E]: row0 = src, row1 = dst
    const int*   bat = (const int*)d_in[2];
    const float* w1[3] = {(const float*)d_in[3],  (const float*)d_in[7],  (const float*)d_in[11]};
    const float* b1[3] = {(const float*)d_in[4],  (const float*)d_in[8],  (const float*)d_in[12]};
    const float* w2[3] = {(const float*)d_in[5],  (const float*)d_in[9],  (const float*)d_in[13]};
    const float* b2[3] = {(const float*)d_in[6],  (const float*)d_in[10], (const float*)d_in[14]};
    const float* fcw = (const float*)d_in[15];
    const float* fcb = (const float*)d_in[16];
    float* out = (float*)d_out;

    const int N = in_sizes[0] / D;
    const int E = in_sizes[1] / 2;
    const int G = out_size;

    const long long nd = (long long)N * D;
    float* buf0   = (float*)d_ws;
    float* buf1   = buf0 + nd;
    float* pooled = buf1 + nd;

    const int*  src = ei;
    const int*  dst = ei + E;

    dim3 scB(256), scG((unsigned)((16LL * E + 255) / 256));
    dim3 gmB(128), gmG((unsigned)((N + 63) / 64));
    dim3 plB(256), plG((unsigned)((16LL * N + 255) / 256));

    // ---- layer 0: h = x ----
    hipMemcpyAsync(buf0, x, nd * sizeof(float), hipMemcpyDeviceToDevice, stream);
    gin_scatter<<<scG, scB, 0, stream>>>(x, buf0, src, dst, E);
    gin_mlp_gemm<<<gmG, gmB, 0, stream>>>(buf0, buf0, w1[0], b1[0], N, 1);
    gin_mlp_gemm<<<gmG, gmB, 0, stream>>>(buf0, buf0, w2[0], b2[0], N, 1); // inter-layer ReLU fused

    // ---- layer 1: h = buf0 ----
    hipMemcpyAsync(buf1, buf0, nd * sizeof(float), hipMemcpyDeviceToDevice, stream);
    gin_scatter<<<scG, scB, 0, stream>>>(buf0, buf1, src, dst, E);
    gin_mlp_gemm<<<gmG, gmB, 0, stream>>>(buf1, buf1, w1[1], b1[1], N, 1);
    gin_mlp_gemm<<<gmG, gmB, 0, stream>>>(buf1, buf1, w2[1], b2[1], N, 1); // inter-layer ReLU fused

    // ---- layer 2: h = buf1 ----
    hipMemcpyAsync(buf0, buf1, nd * sizeof(float), hipMemcpyDeviceToDevice, stream);
    gin_scatter<<<scG, scB, 0, stream>>>(buf1, buf0, src, dst, E);
    gin_mlp_gemm<<<gmG, gmB, 0, stream>>>(buf0, buf0, w1[2], b1[2], N, 1);
    gin_mlp_gemm<<<gmG, gmB, 0, stream>>>(buf0, buf0, w2[2], b2[2], N, 0); // no ReLU after layer 2

    // ---- pool + FC ----
    hipMemsetAsync(pooled, 0, (long long)G * D * sizeof(float), stream);
    gin_pool<<<plG, plB, 0, stream>>>(buf0, bat, pooled, N);
    gin_fc<<<(G + 255) / 256, 256, 0, stream>>>(pooled, fcw, fcb, out, G);
}